// LPN_34918084116809
// MI455X (gfx1250) — compile-verified
//
#include <hip/hip_runtime.h>
#include <hip/hip_bf16.h>
#include <math.h>

// Problem constants (match reference)
#define H_        512
#define W_        512
#define C_        256
#define HW        (H_ * W_)
#define TOPK      4096
#define MAXOUT    512
#define NBINS     4096
#define MIN_SCOREF 0.2f
#define NMS_THR2  64.0f   // 8.0^2

typedef __attribute__((ext_vector_type(2))) float v2f;
typedef __attribute__((ext_vector_type(8))) float v8f;

// ---------------------------------------------------------------------------
// Kernel 1: fused 1x1-conv head via V_WMMA_F32_16X16X4_F32.
// Each wave computes a 16-pixel x 16-col tile (cols 0..2 = logit, reg_y, reg_x;
// cols 3..15 are zero-padded weights). K-loop of 64 steps over 256 channels.
// A-operand (16x4 f32): lane L<16 holds A[M=L][K=0..1] in v0/v1; lane L>=16
// holds A[M=L-16][K=2..3]  -> one contiguous float2 load per lane per step.
// B-operand (4x16 f32): lane holds W[k][n=lane&15], broadcast from LDS.
// Bandwidth-bound: streams 256 MB of feature @ 23.3 TB/s => ~11 us floor.
// ---------------------------------------------------------------------------
__global__ __launch_bounds__(256) void conv_head_wmma(
    const float* __restrict__ feat,
    const float* __restrict__ w_logits, const float* __restrict__ b_logits,
    const float* __restrict__ w_regs,   const float* __restrict__ b_regs,
    float* __restrict__ scores, float* __restrict__ locs)
{
    __shared__ float Wlds[C_][4];        // [k][n], n: 0=logit 1=reg_y 2=reg_x 3=zero
    __shared__ float tile[8][16][16];    // per-wave 16x16 D tile

    const int tid = threadIdx.x;
    for (int i = tid; i < C_; i += 256) {
        Wlds[i][0] = w_logits[i];
        Wlds[i][1] = w_regs[2 * i + 0];
        Wlds[i][2] = w_regs[2 * i + 1];
        Wlds[i][3] = 0.0f;
    }
    __syncthreads();

    const int wave = tid >> 5;
    const int lane = tid & 31;
    const int hl   = lane >> 4;           // 0: K=0..1, 1: K=2..3 half
    const int l16  = lane & 15;           // pixel row M / output column N
    const int pixBase = blockIdx.x * 128 + wave * 16;
    const int ncol = (l16 < 3) ? l16 : 3; // col 3 is the zero pad

    const float* frow = feat + (size_t)(pixBase + l16) * C_ + 2 * hl;

    v8f acc = {};
    for (int k0 = 0; k0 < C_; k0 += 4) {
        if ((k0 & 63) == 0)
            __builtin_prefetch(frow + k0 + 128, 0, 3);   // near-scope prefetch
        v2f a = *(const v2f*)(frow + k0);                 // channels k0+2*hl, +1
        v2f b;
        b.x = Wlds[k0 + 2 * hl + 0][ncol];
        b.y = Wlds[k0 + 2 * hl + 1][ncol];
        acc = __builtin_amdgcn_wmma_f32_16x16x4_f32(
            false, a, false, b, (short)0, acc, false, false);
    }

    // D layout: lane holds N=l16 for M = r + 8*hl  -> scatter to LDS tile
#pragma unroll
    for (int r = 0; r < 8; ++r)
        tile[wave][r + 8 * hl][l16] = acc[r];
    __syncthreads();

    if (hl == 0) {
        const int p   = pixBase + l16;
        float logit   = tile[wave][l16][0] + b_logits[0];
        float ry      = tile[wave][l16][1] + b_regs[0];
        float rx      = tile[wave][l16][2] + b_regs[1];
        // native rcp: 1-ulp sigmoid, avoids the IEEE div expansion
        float s       = __builtin_amdgcn_rcpf(1.0f + __expf(-logit));
        float py      = (float)(p >> 9);      // W_ == 512
        float px      = (float)(p & 511);
        scores[p]     = s;
        locs[2 * p + 0] = (py + 0.5f + ry) * 4.0f;  // FEATURE_SCALE
        locs[2 * p + 1] = (px + 0.5f + rx) * 4.0f;
    }
}

// ---------------------------------------------------------------------------
// Selection pipeline: histogram -> threshold -> compaction -> sort + NMS
// ---------------------------------------------------------------------------
__global__ void zero_ws_kernel(int* hist, int* count)
{
    int i = blockIdx.x * blockDim.x + threadIdx.x;
    if (i < NBINS) hist[i] = 0;
    if (i == 0)    *count = 0;
}

__global__ void hist_kernel(const float* __restrict__ scores, int* __restrict__ hist)
{
    int i = blockIdx.x * blockDim.x + threadIdx.x;
    float s = scores[i];                     // sigmoid output in (0,1)
    int b = (int)(s * (float)NBINS);
    b = b < 0 ? 0 : (b > NBINS - 1 ? NBINS - 1 : b);
    atomicAdd(&hist[b], 1);
}

__global__ void thresh_kernel(const int* __restrict__ hist, float* __restrict__ thr)
{
    if (threadIdx.x == 0 && blockIdx.x == 0) {
        int cum = 0, bin = 0;
        for (int b = NBINS - 1; b >= 0; --b) {
            cum += hist[b]; bin = b;
            if (cum >= TOPK) break;
        }
        *thr = (float)bin / (float)NBINS;    // lower edge of cutoff bin
    }
}

__global__ void compact_kernel(const float* __restrict__ scores,
                               const float* __restrict__ thr,
                               int* __restrict__ count,
                               int* __restrict__ cidx, float* __restrict__ cscore)
{
    int i = blockIdx.x * blockDim.x + threadIdx.x;
    float s = scores[i];
    if (s >= *thr) {
        int slot = atomicAdd(count, 1);
        if (slot < TOPK) { cidx[slot] = i; cscore[slot] = s; }
    }
}

// Single-workgroup: bitonic sort (descending by score) of 4096 candidates in
// LDS, then greedy sequential-pivot NMS. LDS: 16K key + 16K idx(->y) + 16K x
// + 4K active = ~52 KB.
__global__ __launch_bounds__(1024) void sort_nms_kernel(
    const int* __restrict__ count, const int* __restrict__ cidx,
    const float* __restrict__ cscore, const float* __restrict__ locs,
    float* __restrict__ out)
{
    __shared__ float skey[TOPK];
    __shared__ int   sidx[TOPK];          // reused as y-coords after sort
    __shared__ float sxf[TOPK];
    __shared__ unsigned char act[TOPK];
    __shared__ int   nsel;
    __shared__ int   selflag;

    const int tid = threadIdx.x;
    const int cc  = *count;
    const int n   = cc < TOPK ? cc : TOPK;

    for (int j = tid; j < TOPK; j += 1024) {
        if (j < n) { skey[j] = cscore[j]; sidx[j] = cidx[j]; }
        else       { skey[j] = -1.0f;     sidx[j] = -1; }
        act[j] = 1;
    }
    for (int j = tid; j < MAXOUT * 3; j += 1024) out[j] = -1.0f;  // pad output
    if (tid == 0) nsel = 0;
    __syncthreads();

    // Bitonic sort, descending by score
    for (int k = 2; k <= TOPK; k <<= 1) {
        for (int j = k >> 1; j > 0; j >>= 1) {
            for (int t = tid; t < TOPK; t += 1024) {
                int ixj = t ^ j;
                if (ixj > t) {
                    float a = skey[t], b = skey[ixj];
                    bool up = ((t & k) != 0);
                    if ((a > b) == up) {
                        skey[t] = b; skey[ixj] = a;
                        int ti = sidx[t]; sidx[t] = sidx[ixj]; sidx[ixj] = ti;
                    }
                }
            }
            __syncthreads();
        }
    }

    // Gather candidate coordinates into registers (4 slots per thread)
    float jy[4], jx[4];
#pragma unroll
    for (int t = 0; t < 4; ++t) {
        int j  = tid + t * 1024;
        int gi = sidx[j];
        if (gi >= 0) { jy[t] = locs[2 * gi + 0]; jx[t] = locs[2 * gi + 1]; }
        else         { jy[t] = -1.0e9f;          jx[t] = -1.0e9f; }
    }
    __syncthreads();
    float* syf = (float*)sidx;            // overlay: indices no longer needed
#pragma unroll
    for (int t = 0; t < 4; ++t) {
        int j = tid + t * 1024;
        syf[j] = jy[t]; sxf[j] = jx[t];
    }
    __syncthreads();

    // Greedy NMS in score order
    for (int i = 0; i < TOPK; ++i) {
        if (tid == 0) {
            int s = (act[i] && (skey[i] > MIN_SCOREF)) ? 1 : 0;
            selflag = s;
            if (s) {
                out[nsel * 3 + 0] = syf[i];
                out[nsel * 3 + 1] = sxf[i];
                out[nsel * 3 + 2] = skey[i];
                nsel++;
            }
        }
        __syncthreads();
        if (selflag) {
            float piy = syf[i], pix_ = sxf[i];   // LDS broadcast
#pragma unroll
            for (int t = 0; t < 4; ++t) {
                int j = tid + t * 1024;
                float dy = jy[t] - piy, dx = jx[t] - pix_;
                if (dy * dy + dx * dx < NMS_THR2) act[j] = 0;
            }
        }
        __syncthreads();
        if (nsel >= MAXOUT) break;        // uniform: nsel stable post-barrier
    }
}

// ---------------------------------------------------------------------------
extern "C" void kernel_launch(void* const* d_in, const int* in_sizes, int n_in,
                              void* d_out, int out_size, void* d_ws, size_t ws_size,
                              hipStream_t stream)
{
    const float* feat     = (const float*)d_in[0];
    const float* w_logits = (const float*)d_in[1];
    const float* b_logits = (const float*)d_in[2];
    const float* w_regs   = (const float*)d_in[3];
    const float* b_regs   = (const float*)d_in[4];
    float* out = (float*)d_out;

    // Workspace layout (~3.1 MB total)
    char*  ws     = (char*)d_ws;
    float* scores = (float*)ws;  ws += (size_t)HW * sizeof(float);
    float* locs   = (float*)ws;  ws += (size_t)HW * 2 * sizeof(float);
    int*   hist   = (int*)ws;    ws += NBINS * sizeof(int);
    int*   count  = (int*)ws;    ws += 256;
    float* thr    = (float*)ws;  ws += 256;
    int*   cidx   = (int*)ws;    ws += TOPK * sizeof(int);
    float* cscore = (float*)ws;

    conv_head_wmma<<<HW / 128, 256, 0, stream>>>(feat, w_logits, b_logits,
                                                 w_regs, b_regs, scores, locs);
    zero_ws_kernel<<<(NBINS + 255) / 256, 256, 0, stream>>>(hist, count);
    hist_kernel<<<HW / 256, 256, 0, stream>>>(scores, hist);
    thresh_kernel<<<1, 32, 0, stream>>>(hist, thr);
    compact_kernel<<<HW / 256, 256, 0, stream>>>(scores, thr, count, cidx, cscore);
    sort_nms_kernel<<<1, 1024, 0, stream>>>(count, cidx, cscore, locs, out);
}